// ConvOnTree_14474039787898
// MI455X (gfx1250) — compile-verified
//
#include <hip/hip_runtime.h>
#include <hip/hip_bf16.h>

typedef __attribute__((ext_vector_type(2))) float v2f;
typedef __attribute__((ext_vector_type(8))) float v8f;

#define KNB  81   // neighbors
#define CIN  4    // channels per neighbor (x,y,z,dist)
#define COUT 8    // output channels

// One wave handles a tile of 16 points. For each neighbor j (K-chunk of 4):
//   A[16x4] = per-point features (x,y,z,dist), B[4x16] = fused dw*weight (cols 8..15 = 0)
//   D += A x B  via v_wmma_f32_16x16x4_f32
__global__ __launch_bounds__(256)
void conv_tree_wmma(const float* __restrict__ points,
                    const unsigned int* __restrict__ idx32,   // low dwords of the int64 indices
                    const float* __restrict__ dw,             // [K, C]
                    const float* __restrict__ weight,         // [C, K, OUT]
                    const float* __restrict__ bias,           // [OUT]
                    float* __restrict__ out,                  // [N, OUT]
                    int npts, int n_tiles)
{
    __shared__ float w2[KNB * 4 * 16];   // [j][c][l], l padded 8->16 with zeros (20736 B)

    const int tid = threadIdx.x;

    // Build fused weights W2[j][c][l] = dw[j,c] * weight[c,j,l]
    for (int i = tid; i < KNB * 4 * 16; i += 256) {
        const int j = i >> 6;          // / 64
        const int c = (i >> 4) & 3;
        const int l = i & 15;
        float v = 0.0f;
        if (l < COUT) v = dw[j * CIN + c] * weight[(c * KNB + j) * COUT + l];
        w2[i] = v;
    }
    __syncthreads();

    const int lane   = tid & 31;
    const int lane16 = lane & 15;
    const bool hi    = lane >= 16;              // lanes 16..31: K-rows {2,3} of A/B

    // LDS offset of this lane's B-fragment element for row (hi?2:0); +16 floats -> next row
    const int ldsOff0 = (hi ? 2 * 16 : 0) + lane16;

    const int waveGlobal = blockIdx.x * 8 + (tid >> 5);
    const int waveStride = gridDim.x * 8;

    const float bval = bias[lane16 & 7];

    for (int tile = waveGlobal; tile < n_tiles; tile += waveStride) {
        int n = tile * 16 + lane16;
        if (n >= npts) n = npts - 1;            // clamp (keeps EXEC all-1s for WMMA)

        const unsigned int* ip = idx32 + (size_t)n * (2 * KNB);

        v8f acc;
        #pragma unroll
        for (int v = 0; v < 8; ++v) acc[v] = bval;

        // ---- j = 0: the self neighbor (indices[:,0] == n), dist = 1 ----
        const unsigned int idx0 = ip[0];
        const float* p0p = points + (size_t)idx0 * 3;
        const float p0x = p0p[0], p0y = p0p[1], p0z = p0p[2];
        {
            v2f a, b;
            a[0] = hi ? p0z : p0x;
            a[1] = hi ? 1.0f : p0y;
            b[0] = w2[ldsOff0];
            b[1] = w2[ldsOff0 + 16];
            acc = __builtin_amdgcn_wmma_f32_16x16x4_f32(
                false, a, false, b, (short)0, acc, false, false);
        }

        // ---- j = 1 .. 80, 4 neighbors per step: keep 4 gathers in flight ----
        #pragma unroll 1
        for (int j = 1; j < KNB; j += 4) {
            // batch the 4 index loads
            const unsigned int i0 = ip[2 * j + 0];
            const unsigned int i1 = ip[2 * j + 2];
            const unsigned int i2 = ip[2 * j + 4];
            const unsigned int i3 = ip[2 * j + 6];

            // batch the 4 coordinate gathers into distinct registers
            const float* q0 = points + (size_t)i0 * 3;
            const float* q1 = points + (size_t)i1 * 3;
            const float* q2 = points + (size_t)i2 * 3;
            const float* q3 = points + (size_t)i3 * 3;
            const float x0 = q0[0], y0 = q0[1], z0 = q0[2];
            const float x1 = q1[0], y1 = q1[1], z1 = q1[2];
            const float x2 = q2[0], y2 = q2[1], z2 = q2[2];
            const float x3 = q3[0], y3 = q3[1], z3 = q3[2];

            const float dx0 = x0 - p0x, dy0 = y0 - p0y, dz0 = z0 - p0z;
            const float dx1 = x1 - p0x, dy1 = y1 - p0y, dz1 = z1 - p0z;
            const float dx2 = x2 - p0x, dy2 = y2 - p0y, dz2 = z2 - p0z;
            const float dx3 = x3 - p0x, dy3 = y3 - p0y, dz3 = z3 - p0z;
            const float d0 = dx0 * dx0 + dy0 * dy0 + dz0 * dz0 + 1.0f;
            const float d1 = dx1 * dx1 + dy1 * dy1 + dz1 * dz1 + 1.0f;
            const float d2 = dx2 * dx2 + dy2 * dy2 + dz2 * dz2 + 1.0f;
            const float d3 = dx3 * dx3 + dy3 * dy3 + dz3 * dz3 + 1.0f;

            v2f a0, a1, a2, a3, b0, b1, b2, b3;
            a0[0] = hi ? z0 : x0;  a0[1] = hi ? d0 : y0;
            a1[0] = hi ? z1 : x1;  a1[1] = hi ? d1 : y1;
            a2[0] = hi ? z2 : x2;  a2[1] = hi ? d2 : y2;
            a3[0] = hi ? z3 : x3;  a3[1] = hi ? d3 : y3;

            const int wbase = j * 64 + ldsOff0;
            b0[0] = w2[wbase +   0];  b0[1] = w2[wbase +  16];
            b1[0] = w2[wbase +  64];  b1[1] = w2[wbase +  80];
            b2[0] = w2[wbase + 128];  b2[1] = w2[wbase + 144];
            b3[0] = w2[wbase + 192];  b3[1] = w2[wbase + 208];

            acc = __builtin_amdgcn_wmma_f32_16x16x4_f32(
                false, a0, false, b0, (short)0, acc, false, false);
            acc = __builtin_amdgcn_wmma_f32_16x16x4_f32(
                false, a1, false, b1, (short)0, acc, false, false);
            acc = __builtin_amdgcn_wmma_f32_16x16x4_f32(
                false, a2, false, b2, (short)0, acc, false, false);
            acc = __builtin_amdgcn_wmma_f32_16x16x4_f32(
                false, a3, false, b3, (short)0, acc, false, false);
        }

        // D layout: lane L, vgpr v -> row M = v + (hi?8:0), col N = lane16.
        if (lane16 < COUT) {
            #pragma unroll
            for (int v = 0; v < 8; ++v) {
                const int row = tile * 16 + v + (hi ? 8 : 0);
                if (row < npts) out[(size_t)row * COUT + lane16] = acc[v];
            }
        }
    }
}

extern "C" void kernel_launch(void* const* d_in, const int* in_sizes, int n_in,
                              void* d_out, int out_size, void* d_ws, size_t ws_size,
                              hipStream_t stream) {
    const float*        points = (const float*)d_in[0];         // [N,3] f32
    const unsigned int* idx32  = (const unsigned int*)d_in[1];  // [N,K] int64, low dwords used
    const float*        dw     = (const float*)d_in[2];         // [K,C] f32
    const float*        weight = (const float*)d_in[3];         // [C,K,OUT] f32
    const float*        bias   = (const float*)d_in[4];         // [OUT] f32
    float*              out    = (float*)d_out;                 // [N,OUT] f32

    const int npts    = in_sizes[0] / 3;
    const int n_tiles = (npts + 15) / 16;

    int blocks = (n_tiles + 7) / 8;     // 8 waves per block, 1 tile per wave per pass
    if (blocks > 2048) blocks = 2048;   // grid-stride over remaining tiles
    if (blocks < 1) blocks = 1;

    conv_tree_wmma<<<blocks, 256, 0, stream>>>(points, idx32, dw, weight, bias,
                                               out, npts, n_tiles);
}